// GATNet_56418690400272
// MI455X (gfx1250) — compile-verified
//
#include <hip/hip_runtime.h>
#include <math.h>

#define DIMC 128
#define NEG_SLOPE 0.2f
#define LN_EPSF 1e-5f
#define SM_EPSF 1e-16f

typedef float v2f __attribute__((ext_vector_type(2)));
typedef float v8f __attribute__((ext_vector_type(8)));

__device__ __forceinline__ float lrelu(float x) {
    return x > 0.f ? x : NEG_SLOPE * x;
}

// float atomic max via sign-magnitude ordering trick (init value must be -inf bits)
__device__ __forceinline__ void atomicMaxFloat(float* addr, float val) {
    if (val >= 0.f) atomicMax((int*)addr, __float_as_int(val));
    else            atomicMin((unsigned int*)addr, __float_as_uint(val));
}

__device__ __forceinline__ void atomicAddF(float* addr, float val) {
    unsafeAtomicAdd(addr, val);   // native global_atomic_add_f32
}

// ---------------------------------------------------------------- fill
__global__ void fill_kernel(float* __restrict__ p, float v, long long n) {
    long long i = (long long)blockIdx.x * blockDim.x + threadIdx.x;
    if (i < n) p[i] = v;
}

// ---------------------------------------------------------------- LayerNorm (wave per node)
__global__ __launch_bounds__(256)
void ln_kernel(const float* __restrict__ x, const float* __restrict__ g,
               const float* __restrict__ b, float* __restrict__ out, int nn) {
    int wave = threadIdx.x >> 5, lane = threadIdx.x & 31;
    int node = blockIdx.x * 8 + wave;
    if (node >= nn) return;
    float4 v = ((const float4*)(x + (long long)node * DIMC))[lane];
    float s = v.x + v.y + v.z + v.w;
#pragma unroll
    for (int off = 16; off; off >>= 1) s += __shfl_xor(s, off, 32);
    float mu = s * (1.f / DIMC);
    float dx = v.x - mu, dy = v.y - mu, dz = v.z - mu, dw = v.w - mu;
    float q = dx * dx + dy * dy + dz * dz + dw * dw;
#pragma unroll
    for (int off = 16; off; off >>= 1) q += __shfl_xor(q, off, 32);
    float rs = rsqrtf(q * (1.f / DIMC) + LN_EPSF);
    float4 gg = ((const float4*)g)[lane];
    float4 bb = ((const float4*)b)[lane];
    float4 o;
    o.x = dx * rs * gg.x + bb.x;
    o.y = dy * rs * gg.y + bb.y;
    o.z = dz * rs * gg.z + bb.z;
    o.w = dw * rs * gg.w + bb.w;
    ((float4*)(out + (long long)node * DIMC))[lane] = o;
}

// ---------------------------------------------------------------- dual GEMM via fp32 WMMA
// xl = h @ Wl, xr = h @ Wr ; h:[M,128], W:[128,128] row-major. M % 16 == 0.
// Block: 256 threads = 8 waves. Wave w owns output cols [16w,16w+16). K-loop 128/4.
__global__ __launch_bounds__(256)
void gemm_dual_kernel(const float* __restrict__ h,
                      const float* __restrict__ Wl,
                      const float* __restrict__ Wr,
                      float* __restrict__ xl, float* __restrict__ xr) {
    __shared__ float Ash[16][132];           // pad stride to kill bank conflicts
    const int row0 = blockIdx.x * 16;
    const int t = threadIdx.x;
    {   // cooperative stage of 16x128 A tile: 8 floats per thread
        int r = t >> 4, c0 = (t & 15) * 8;
        const float* src = h + (long long)(row0 + r) * DIMC + c0;
        float4 p0 = ((const float4*)src)[0];
        float4 p1 = ((const float4*)src)[1];
        Ash[r][c0 + 0] = p0.x; Ash[r][c0 + 1] = p0.y;
        Ash[r][c0 + 2] = p0.z; Ash[r][c0 + 3] = p0.w;
        Ash[r][c0 + 4] = p1.x; Ash[r][c0 + 5] = p1.y;
        Ash[r][c0 + 6] = p1.z; Ash[r][c0 + 7] = p1.w;
    }
    __syncthreads();

    const int lane = t & 31;
    const int col0 = (t >> 5) * 16;
    const int n    = lane & 15;              // output column within tile
    const int r    = lane & 15;              // A row held by this lane
    const int koff = (lane < 16) ? 0 : 2;    // K sub-offset per ISA fp32 A/B layout

    v8f cl = {0.f, 0.f, 0.f, 0.f, 0.f, 0.f, 0.f, 0.f};
    v8f cr = {0.f, 0.f, 0.f, 0.f, 0.f, 0.f, 0.f, 0.f};

#pragma unroll
    for (int kb = 0; kb < DIMC; kb += 4) {
        v2f a;
        a.x = Ash[r][kb + koff];
        a.y = Ash[r][kb + koff + 1];
        v2f bl, br;
        bl.x = Wl[(long long)(kb + koff) * DIMC + col0 + n];
        bl.y = Wl[(long long)(kb + koff + 1) * DIMC + col0 + n];
        br.x = Wr[(long long)(kb + koff) * DIMC + col0 + n];
        br.y = Wr[(long long)(kb + koff + 1) * DIMC + col0 + n];
        cl = __builtin_amdgcn_wmma_f32_16x16x4_f32(false, a, false, bl, (short)0, cl, false, false);
        cr = __builtin_amdgcn_wmma_f32_16x16x4_f32(false, a, false, br, (short)0, cr, false, false);
    }

    const int mbase = (lane < 16) ? 0 : 8;   // C/D layout: vgpr i = rows i / i+8
#pragma unroll
    for (int i = 0; i < 8; ++i) {
        long long orow = (long long)(row0 + mbase + i) * DIMC + col0 + n;
        xl[orow] = cl[i];
        xr[orow] = cr[i];
    }
}

// ---------------------------------------------------------------- edge scoring (wave per edge)
template <int H>
__global__ __launch_bounds__(256)
void edge_score_kernel(const float* __restrict__ xl, const float* __restrict__ xr,
                       const int* __restrict__ ei, const float* __restrict__ att,
                       float* __restrict__ esc, float* __restrict__ nmax, int E_) {
    const int wave = threadIdx.x >> 5, lane = threadIdx.x & 31;
    const long long e = (long long)blockIdx.x * 8 + wave;
    if (e >= E_) return;
    const int src = ei[e];
    const int dst = ei[(long long)E_ + e];
    float4 ml = ((const float4*)(xl + (long long)src * DIMC))[lane];
    float4 mr = ((const float4*)(xr + (long long)dst * DIMC))[lane];
    float4 a4 = ((const float4*)att)[lane];
    float s = 0.f, m;
    m = ml.x + mr.x; s += lrelu(m) * a4.x;
    m = ml.y + mr.y; s += lrelu(m) * a4.y;
    m = ml.z + mr.z; s += lrelu(m) * a4.z;
    m = ml.w + mr.w; s += lrelu(m) * a4.w;
    const int G = 32 / H;                    // lanes per head
#pragma unroll
    for (int off = G >> 1; off; off >>= 1) s += __shfl_xor(s, off, 32);
    if ((lane & (G - 1)) == 0) {
        int hh = lane / G;
        esc[e * H + hh] = s;
        atomicMaxFloat(&nmax[(long long)dst * H + hh], s);
    }
}

// ---------------------------------------------------------------- exp + denom (thread per edge*head)
template <int H>
__global__ __launch_bounds__(256)
void edge_exp_kernel(const int* __restrict__ ei, float* __restrict__ esc,
                     const float* __restrict__ nmax, float* __restrict__ den, int E_) {
    long long i = (long long)blockIdx.x * blockDim.x + threadIdx.x;
    if (i >= (long long)E_ * H) return;
    long long e = i / H;
    int hh = (int)(i - e * H);
    int dst = ei[(long long)E_ + e];
    float ex = expf(esc[i] - nmax[(long long)dst * H + hh]);
    esc[i] = ex;
    atomicAddF(&den[(long long)dst * H + hh], ex);
}

// ---------------------------------------------------------------- weighted scatter (wave per edge)
template <int H>
__global__ __launch_bounds__(256)
void edge_aggr_kernel(const float* __restrict__ xl, const int* __restrict__ ei,
                      const float* __restrict__ esc, const float* __restrict__ den,
                      float* __restrict__ acc, int E_) {
    const int wave = threadIdx.x >> 5, lane = threadIdx.x & 31;
    const long long e = (long long)blockIdx.x * 8 + wave;
    if (e >= E_) return;
    const int src = ei[e];
    const int dst = ei[(long long)E_ + e];
    const int G = 32 / H;
    const int hh = lane / G;
    float alpha = esc[e * H + hh] / (den[(long long)dst * H + hh] + SM_EPSF);
    float4 v = ((const float4*)(xl + (long long)src * DIMC))[lane];
    float* o = acc + (long long)dst * DIMC + lane * 4;
    atomicAddF(o + 0, alpha * v.x);
    atomicAddF(o + 1, alpha * v.y);
    atomicAddF(o + 2, alpha * v.z);
    atomicAddF(o + 3, alpha * v.w);
}

// ---------------------------------------------------------------- acc + b0 + x -> GELU -> LN
__global__ __launch_bounds__(256)
void post0_kernel(const float* __restrict__ acc, const float* __restrict__ bias,
                  const float* __restrict__ x, const float* __restrict__ g,
                  const float* __restrict__ b, float* __restrict__ out, int nn) {
    int wave = threadIdx.x >> 5, lane = threadIdx.x & 31;
    int node = blockIdx.x * 8 + wave;
    if (node >= nn) return;
    float4 a = ((const float4*)(acc + (long long)node * DIMC))[lane];
    float4 bi = ((const float4*)bias)[lane];
    float4 xv = ((const float4*)(x + (long long)node * DIMC))[lane];
    float t0 = a.x + bi.x + xv.x, t1 = a.y + bi.y + xv.y;
    float t2 = a.z + bi.z + xv.z, t3 = a.w + bi.w + xv.w;
    const float k = 0.70710678118654752440f;
    float u0 = 0.5f * t0 * (1.f + erff(t0 * k));
    float u1 = 0.5f * t1 * (1.f + erff(t1 * k));
    float u2 = 0.5f * t2 * (1.f + erff(t2 * k));
    float u3 = 0.5f * t3 * (1.f + erff(t3 * k));
    float s = u0 + u1 + u2 + u3;
#pragma unroll
    for (int off = 16; off; off >>= 1) s += __shfl_xor(s, off, 32);
    float mu = s * (1.f / DIMC);
    float d0 = u0 - mu, d1 = u1 - mu, d2 = u2 - mu, d3 = u3 - mu;
    float q = d0 * d0 + d1 * d1 + d2 * d2 + d3 * d3;
#pragma unroll
    for (int off = 16; off; off >>= 1) q += __shfl_xor(q, off, 32);
    float rs = rsqrtf(q * (1.f / DIMC) + LN_EPSF);
    float4 gg = ((const float4*)g)[lane];
    float4 bb = ((const float4*)b)[lane];
    float4 o;
    o.x = d0 * rs * gg.x + bb.x;
    o.y = d1 * rs * gg.y + bb.y;
    o.z = d2 * rs * gg.z + bb.z;
    o.w = d3 * rs * gg.w + bb.w;
    ((float4*)(out + (long long)node * DIMC))[lane] = o;
}

// ---------------------------------------------------------------- out = acc + b1 + x
__global__ void final_kernel(const float* __restrict__ acc, const float* __restrict__ bias,
                             const float* __restrict__ x, float* __restrict__ out, long long n) {
    long long i = (long long)blockIdx.x * blockDim.x + threadIdx.x;
    if (i < n) out[i] = acc[i] + bias[i & (DIMC - 1)] + x[i];
}

// ================================================================ launch
extern "C" void kernel_launch(void* const* d_in, const int* in_sizes, int n_in,
                              void* d_out, int out_size, void* d_ws, size_t ws_size,
                              hipStream_t stream) {
    const float* x     = (const float*)d_in[0];
    const int*   ei    = (const int*)d_in[1];
    const float* ln0_g = (const float*)d_in[2];
    const float* ln0_b = (const float*)d_in[3];
    const float* W0l   = (const float*)d_in[4];
    const float* W0r   = (const float*)d_in[5];
    const float* att0  = (const float*)d_in[6];
    const float* b0    = (const float*)d_in[7];
    const float* ln1_g = (const float*)d_in[8];
    const float* ln1_b = (const float*)d_in[9];
    const float* W1l   = (const float*)d_in[10];
    const float* W1r   = (const float*)d_in[11];
    const float* att1  = (const float*)d_in[12];
    const float* b1    = (const float*)d_in[13];
    float* out = (float*)d_out;

    const int Nn = in_sizes[0] / DIMC;   // 50000
    const int E  = in_sizes[1] / 2;      // 800000

    float* ws  = (float*)d_ws;
    float* h   = ws;  ws += (long long)Nn * DIMC;
    float* xl  = ws;  ws += (long long)Nn * DIMC;
    float* xr  = ws;  ws += (long long)Nn * DIMC;
    float* esc = ws;  ws += (long long)E * 8;
    float* nmx = ws;  ws += (long long)Nn * 8;
    float* den = ws;  ws += (long long)Nn * 8;
    float* acc = ws;  ws += (long long)Nn * DIMC;

    auto cdiv = [](long long a, long long b) { return (int)((a + b - 1) / b); };
    const long long NF = (long long)Nn * DIMC;

    // ---------------- layer 0 (H=8) ----------------
    fill_kernel<<<cdiv((long long)Nn * 8, 256), 256, 0, stream>>>(nmx, -INFINITY, (long long)Nn * 8);
    fill_kernel<<<cdiv((long long)Nn * 8, 256), 256, 0, stream>>>(den, 0.f, (long long)Nn * 8);
    fill_kernel<<<cdiv(NF, 256), 256, 0, stream>>>(acc, 0.f, NF);
    ln_kernel<<<cdiv(Nn, 8), 256, 0, stream>>>(x, ln0_g, ln0_b, h, Nn);
    gemm_dual_kernel<<<Nn / 16, 256, 0, stream>>>(h, W0l, W0r, xl, xr);
    edge_score_kernel<8><<<cdiv(E, 8), 256, 0, stream>>>(xl, xr, ei, att0, esc, nmx, E);
    edge_exp_kernel<8><<<cdiv((long long)E * 8, 256), 256, 0, stream>>>(ei, esc, nmx, den, E);
    edge_aggr_kernel<8><<<cdiv(E, 8), 256, 0, stream>>>(xl, ei, esc, den, acc, E);
    post0_kernel<<<cdiv(Nn, 8), 256, 0, stream>>>(acc, b0, x, ln1_g, ln1_b, h, Nn);

    // ---------------- layer 1 (H=1) ----------------
    fill_kernel<<<cdiv((long long)Nn, 256), 256, 0, stream>>>(nmx, -INFINITY, (long long)Nn);
    fill_kernel<<<cdiv((long long)Nn, 256), 256, 0, stream>>>(den, 0.f, (long long)Nn);
    fill_kernel<<<cdiv(NF, 256), 256, 0, stream>>>(acc, 0.f, NF);
    gemm_dual_kernel<<<Nn / 16, 256, 0, stream>>>(h, W1l, W1r, xl, xr);
    edge_score_kernel<1><<<cdiv(E, 8), 256, 0, stream>>>(xl, xr, ei, att1, esc, nmx, E);
    edge_exp_kernel<1><<<cdiv((long long)E, 256), 256, 0, stream>>>(ei, esc, nmx, den, E);
    edge_aggr_kernel<1><<<cdiv(E, 8), 256, 0, stream>>>(xl, ei, esc, den, acc, E);
    final_kernel<<<cdiv(NF, 256), 256, 0, stream>>>(acc, b1, x, out, NF);
}